// RGBD_CAM_Module_5729486373127
// MI455X (gfx1250) — compile-verified
//
#include <hip/hip_runtime.h>
#include <stdint.h>

typedef __attribute__((ext_vector_type(2))) float v2f;
typedef __attribute__((ext_vector_type(8))) float v8f;

#define C_DIM 512
#define N_DIM 4096
#define TM 128
#define TN 128
#define TK 32
#define LDS_K 36    // [row][k] stride in floats (16B-aligned rows)
#define LDS_N 132   // [k][n] stride in floats (16B-aligned rows)

// ---------------------------------------------------------------------------
// CDNA5 async global->LDS copy (ASYNCcnt-tracked, no VGPR staging).
// lds_addr: wave-relative LDS byte offset (low 32 bits of generic pointer).
// gaddr:    per-lane 64-bit global address (GV mode, saddr = off).
// ---------------------------------------------------------------------------
__device__ __forceinline__ void async_copy_b128(uint32_t lds_addr, const float* gaddr)
{
    asm volatile("global_load_async_to_lds_b128 %0, %1, off"
                 :: "v"(lds_addr), "v"((uint64_t)(uintptr_t)gaddr)
                 : "memory");
}

__device__ __forceinline__ void wait_async0()
{
    asm volatile("s_wait_asynccnt 0" ::: "memory");
}

// ---------------------------------------------------------------------------
// Kernel 1: S[b] = Q_b Q_b^T + D_b D_b^T   (512x512 per batch, K = 4096)
// Double-buffered async LDS pipeline, f32 WMMA 16x16x4.
// ---------------------------------------------------------------------------
__global__ __launch_bounds__(256, 1)
void gram_kernel(const float* __restrict__ xr,
                 const float* __restrict__ xd,
                 float* __restrict__ S)
{
    __shared__ float As[2][TM][LDS_K];
    __shared__ float Bs[2][TN][LDS_K];

    const int b   = blockIdx.z;
    const int m0  = blockIdx.y * TM;
    const int n0  = blockIdx.x * TN;

    const int tid  = threadIdx.x;
    const int lane = tid & 31;
    const int wave = tid >> 5;
    const int wm   = (wave >> 2) * 64;      // wave M offset: 0 or 64
    const int wn   = (wave & 3) * 32;       // wave N offset: 0,32,64,96
    const int lm   = lane & 15;
    const int khi  = (lane >> 4) << 1;      // K pair base: 0 or 2

    const uint32_t as_base = (uint32_t)(uintptr_t)(void*)&As[0][0][0];
    const uint32_t bs_base = (uint32_t)(uintptr_t)(void*)&Bs[0][0][0];
    const uint32_t BUFB    = TM * LDS_K * 4;     // bytes per buffer

    const float* q0 = xr + (size_t)b * C_DIM * N_DIM;
    const float* q1 = xd + (size_t)b * C_DIM * N_DIM;

    auto prefetch = [&](int buf, const float* q, int k0) {
        for (int r = 0; r < 4; ++r) {
            int idx = tid + r * 256;        // 0..1023
            int row = idx >> 3;             // 0..127
            int c4  = (idx & 7) << 2;       // 0..28
            uint32_t loff = (uint32_t)(row * LDS_K + c4) * 4u + (uint32_t)buf * BUFB;
            async_copy_b128(as_base + loff, q + (size_t)(m0 + row) * N_DIM + k0 + c4);
            async_copy_b128(bs_base + loff, q + (size_t)(n0 + row) * N_DIM + k0 + c4);
        }
    };

    v8f acc[4][2];
    {
        v8f z = {0.f, 0.f, 0.f, 0.f, 0.f, 0.f, 0.f, 0.f};
        for (int i = 0; i < 4; ++i)
            for (int j = 0; j < 2; ++j)
                acc[i][j] = z;
    }

    const int NBLK = 2 * (N_DIM / TK);      // 128 rgb blocks + 128 dep blocks
    prefetch(0, q0, 0);

    for (int blk = 0; blk < NBLK; ++blk) {
        const int cur = blk & 1;
        wait_async0();          // this wave's async writes landed in LDS
        __syncthreads();        // everyone's writes visible; other buffer free

        if (blk + 1 < NBLK) {
            const int nb = blk + 1;
            prefetch(cur ^ 1, (nb < NBLK / 2) ? q0 : q1, (nb & (NBLK / 2 - 1)) * TK);
        }

        for (int kb = 0; kb < TK; kb += 4) {
            const int kk = kb + khi;
            v2f a[4], bb[2];
            for (int i = 0; i < 4; ++i)
                a[i] = *(const v2f*)&As[cur][wm + i * 16 + lm][kk];
            for (int j = 0; j < 2; ++j)
                bb[j] = *(const v2f*)&Bs[cur][wn + j * 16 + lm][kk];
            for (int i = 0; i < 4; ++i)
                for (int j = 0; j < 2; ++j)
                    acc[i][j] = __builtin_amdgcn_wmma_f32_16x16x4_f32(
                        false, a[i], false, bb[j],
                        (short)0, acc[i][j], false, false);
        }
    }

    // C/D layout: VGPR r -> lanes 0-15 row r, lanes 16-31 row r+8.
    const int rof = (lane >> 4) << 3;
    float* Sb = S + (size_t)b * C_DIM * C_DIM;
    for (int i = 0; i < 4; ++i)
        for (int j = 0; j < 2; ++j)
            for (int r = 0; r < 8; ++r) {
                int row = m0 + wm + i * 16 + r + rof;
                int col = n0 + wn + j * 16 + lm;
                Sb[(size_t)row * C_DIM + col] = acc[i][j][r];
            }
}

// ---------------------------------------------------------------------------
// Kernel 2: attention = softmax(-S) row-wise, in place.
// softmax(-S) = exp(minS - S) / sum
// ---------------------------------------------------------------------------
__global__ __launch_bounds__(256, 1)
void softmax_kernel(float* __restrict__ S)
{
    __shared__ float red[256];
    const int row = blockIdx.x;
    float* p = S + (size_t)row * C_DIM;
    const int tid = threadIdx.x;

    float v0 = p[tid];
    float v1 = p[tid + 256];

    red[tid] = fminf(v0, v1);
    __syncthreads();
    for (int s = 128; s > 0; s >>= 1) {
        if (tid < s) red[tid] = fminf(red[tid], red[tid + s]);
        __syncthreads();
    }
    const float mn = red[0];
    __syncthreads();

    float e0 = __expf(mn - v0);
    float e1 = __expf(mn - v1);
    red[tid] = e0 + e1;
    __syncthreads();
    for (int s = 128; s > 0; s >>= 1) {
        if (tid < s) red[tid] += red[tid + s];
        __syncthreads();
    }
    const float inv = 1.0f / red[0];

    p[tid]       = e0 * inv;
    p[tid + 256] = e1 * inv;
}

// ---------------------------------------------------------------------------
// Kernel 3: y = gamma * (attention @ Q) + x_rgb    (M=512, N=4096, K=512)
// A tile [m][k]; B tile kept natural [k][n] (async copy cannot transpose) --
// B fragment = two b32 LDS loads, constant row-stride apart (-> ds_load_2addr).
// ---------------------------------------------------------------------------
__global__ __launch_bounds__(256, 1)
void out_kernel(const float* __restrict__ attn,
                const float* __restrict__ xr,
                const float* __restrict__ gamma,
                float* __restrict__ y)
{
    __shared__ float As[2][TM][LDS_K];   // attention tile [m][k]
    __shared__ float Bs[2][TK][LDS_N];   // Q tile [k][n]

    const int b   = blockIdx.z;
    const int m0  = blockIdx.y * TM;     // over C (4 tiles)
    const int n0  = blockIdx.x * TN;     // over N (32 tiles)

    const float* A = attn + (size_t)b * C_DIM * C_DIM;
    const float* Q = xr   + (size_t)b * C_DIM * N_DIM;

    const int tid  = threadIdx.x;
    const int lane = tid & 31;
    const int wave = tid >> 5;
    const int wm   = (wave >> 2) * 64;
    const int wn   = (wave & 3) * 32;
    const int lm   = lane & 15;
    const int khi  = (lane >> 4) << 1;

    const uint32_t as_base = (uint32_t)(uintptr_t)(void*)&As[0][0][0];
    const uint32_t bs_base = (uint32_t)(uintptr_t)(void*)&Bs[0][0][0];
    const uint32_t ABUFB   = TM * LDS_K * 4;
    const uint32_t BBUFB   = TK * LDS_N * 4;

    auto prefetch = [&](int buf, int k0) {
        for (int r = 0; r < 4; ++r) {
            int idx = tid + r * 256;
            int row = idx >> 3;             // 0..127 (A rows)
            int c4  = (idx & 7) << 2;       // 0..28  (A k-chunk)
            async_copy_b128(as_base + (uint32_t)buf * ABUFB +
                                (uint32_t)(row * LDS_K + c4) * 4u,
                            A + (size_t)(m0 + row) * C_DIM + k0 + c4);
            int krow = idx >> 5;            // 0..31  (B k-rows)
            int n4   = (idx & 31) << 2;     // 0..124 (B n-chunk)
            async_copy_b128(bs_base + (uint32_t)buf * BBUFB +
                                (uint32_t)(krow * LDS_N + n4) * 4u,
                            Q + (size_t)(k0 + krow) * N_DIM + n0 + n4);
        }
    };

    v8f acc[4][2];
    {
        v8f z = {0.f, 0.f, 0.f, 0.f, 0.f, 0.f, 0.f, 0.f};
        for (int i = 0; i < 4; ++i)
            for (int j = 0; j < 2; ++j)
                acc[i][j] = z;
    }

    const int NBLK = C_DIM / TK;            // 16
    prefetch(0, 0);

    for (int blk = 0; blk < NBLK; ++blk) {
        const int cur = blk & 1;
        wait_async0();
        __syncthreads();

        if (blk + 1 < NBLK)
            prefetch(cur ^ 1, (blk + 1) * TK);

        for (int kb = 0; kb < TK; kb += 4) {
            const int kk = kb + khi;
            v2f a[4], bb[2];
            for (int i = 0; i < 4; ++i)
                a[i] = *(const v2f*)&As[cur][wm + i * 16 + lm][kk];
            for (int j = 0; j < 2; ++j) {
                bb[j].x = Bs[cur][kk    ][wn + j * 16 + lm];
                bb[j].y = Bs[cur][kk + 1][wn + j * 16 + lm];
            }
            for (int i = 0; i < 4; ++i)
                for (int j = 0; j < 2; ++j)
                    acc[i][j] = __builtin_amdgcn_wmma_f32_16x16x4_f32(
                        false, a[i], false, bb[j],
                        (short)0, acc[i][j], false, false);
        }
    }

    const float g   = gamma[0];
    const int   rof = (lane >> 4) << 3;
    for (int i = 0; i < 4; ++i)
        for (int j = 0; j < 2; ++j)
            for (int r = 0; r < 8; ++r) {
                int row = m0 + wm + i * 16 + r + rof;
                int col = n0 + wn + j * 16 + lm;
                size_t o = (size_t)row * N_DIM + col;
                y[(size_t)b * C_DIM * N_DIM + o] = g * acc[i][j][r] + Q[o];
            }
}

// ---------------------------------------------------------------------------
extern "C" void kernel_launch(void* const* d_in, const int* in_sizes, int n_in,
                              void* d_out, int out_size, void* d_ws, size_t ws_size,
                              hipStream_t stream)
{
    const float* xr    = (const float*)d_in[0];
    const float* xd    = (const float*)d_in[1];
    const float* gamma = (const float*)d_in[2];
    float* y = (float*)d_out;
    float* S = (float*)d_ws;   // 8 * 512 * 512 floats = 8 MB scratch

    dim3 blk(256, 1, 1);
    gram_kernel   <<<dim3(C_DIM / TN, C_DIM / TM, 8), blk, 0, stream>>>(xr, xd, S);
    softmax_kernel<<<dim3(8 * C_DIM, 1, 1),           blk, 0, stream>>>(S);
    out_kernel    <<<dim3(N_DIM / TN, C_DIM / TM, 8), blk, 0, stream>>>(S, xr, gamma, y);
}